// ReplicateCoherenceLoss_25503515804232
// MI455X (gfx1250) — compile-verified
//
#include <hip/hip_runtime.h>

// Problem constants (from the reference).
#define B_DIM 512
#define T_DIM 4096
#define R_DIM 16

// Partitioning: one block = one batch row b x one T-segment.
#define SPLIT 4                               // T-segments per batch row
#define NB (B_DIM * SPLIT)                    // 2048 blocks -> 2048 partials
#define T_PER_BLOCK (T_DIM / SPLIT)           // 1024 t-rows per block
#define TILES_PER_BLOCK (T_PER_BLOCK / 16)    // 64 16-row tiles per block
#define WAVES_PER_BLOCK 8                     // 256 threads / wave32

typedef __attribute__((ext_vector_type(2))) float v2f;
typedef __attribute__((ext_vector_type(8))) float v8f;

// ---------------------------------------------------------------------------
// Kernel 1: stream pred/target; per 16x16 f32 tile compute, via chained
// V_WMMA_F32_16X16X4_F32 (K=16 as 4 chunks of K=4, B = mask in column 0):
//   S1p = P * m,   S1t = T * m,   S2d = (P oP - T oT) * m
// Then per row:
//   var_p - var_t = (S2d - (S1p-S1t)(S1p+S1t)/n) / (n-1)
// which saves a 4th WMMA chain vs computing S2p and S2t separately.
// A-operand layout (ISA 7.12.2, 32-bit A 16x4): lanes 0-15 hold K={4c,4c+1},
// lanes 16-31 hold K={4c+2,4c+3} for row M=lane%16 -> b64 loads feed WMMA
// directly, no cross-lane movement. Columns N>0 of B are zero, so every lane
// except N==0 accumulates exact 0 and contributes exact 0 to the loss.
// ---------------------------------------------------------------------------
__global__ __launch_bounds__(256) void rc_var_wmma_kernel(
    const float* __restrict__ pred, const float* __restrict__ target,
    const float* __restrict__ mask, float* __restrict__ partial)
{
  __shared__ float s_red[256];
  __shared__ float s_n;

  const int tid  = threadIdx.x;
  const int lane = tid & 31;
  const int wave = tid >> 5;
  const int bid  = blockIdx.x;
  const int b    = bid / SPLIT;
  const int seg  = bid % SPLIT;

  const float* mrow = mask + b * R_DIM;

  // Per-batch mask population count.
  if (tid == 0) {
    float n = 0.f;
    for (int r = 0; r < R_DIM; ++r) n += mrow[r];
    s_n = n;
  }
  __syncthreads();
  const float n       = s_n;
  const float ns      = fmaxf(n, 2.0f);          // n_safe
  const float inv_n   = 1.0f / ns;
  const float inv_nm1 = 1.0f / (ns - 1.0f);
  const float valid   = (n > 1.0f) ? 1.0f : 0.0f;

  // Lane's element offset inside a 16x16 tile (floats):
  // row = lane%16, K-half = lane/16 selects columns {+0,+1} vs {+2,+3}.
  const int laneOff = (lane & 15) * R_DIM + (lane >> 4) * 2;

  // Build the 4 B-operand chunks once: column 0 of B = mask (same K split
  // as the A layout: first VGPR = K {4c | 4c+2}, second = K {4c+1 | 4c+3}).
  v2f bm[4];
  {
    const int koff = (lane >> 4) * 2;
#pragma unroll
    for (int c = 0; c < 4; ++c) {
      v2f mv; mv.x = mrow[4 * c + koff + 0];
              mv.y = mrow[4 * c + koff + 1];
      v2f z;  z.x = 0.f; z.y = 0.f;
      bm[c] = ((lane & 15) == 0) ? mv : z;     // only column N==0 nonzero
    }
  }

  const size_t baseRow =
      ((size_t)b * T_DIM + (size_t)seg * T_PER_BLOCK) * R_DIM;

  float acc = 0.0f;

  for (int t = wave; t < TILES_PER_BLOCK; t += WAVES_PER_BLOCK) {
    const float* tp = pred   + baseRow + (size_t)t * (16 * R_DIM) + laneOff;
    const float* tt = target + baseRow + (size_t)t * (16 * R_DIM) + laneOff;

    // Prefetch this wave's next tile into GL2 (speculative; dropped if OOB).
    __builtin_prefetch(tp + WAVES_PER_BLOCK * 16 * R_DIM, 0, 0);
    __builtin_prefetch(tt + WAVES_PER_BLOCK * 16 * R_DIM, 0, 0);

    v8f dS1p = {}; v8f dS1t = {}; v8f dSd = {};
#pragma unroll
    for (int c = 0; c < 4; ++c) {
      // Streaming (read-once) data: non-temporal so we don't thrash L2.
      v2f pa = __builtin_nontemporal_load((const v2f*)(tp + 4 * c));
      v2f ta = __builtin_nontemporal_load((const v2f*)(tt + 4 * c));
      // d2 = p*p - t*t  (one pk_mul + one pk_fma with neg)
      v2f d2;
      d2.x = fmaf(pa.x, pa.x, -(ta.x * ta.x));
      d2.y = fmaf(pa.y, pa.y, -(ta.y * ta.y));
      dS1p = __builtin_amdgcn_wmma_f32_16x16x4_f32(false, pa, false, bm[c],
                                                   (short)0, dS1p, false, false);
      dS1t = __builtin_amdgcn_wmma_f32_16x16x4_f32(false, ta, false, bm[c],
                                                   (short)0, dS1t, false, false);
      dSd  = __builtin_amdgcn_wmma_f32_16x16x4_f32(false, d2, false, bm[c],
                                                   (short)0, dSd,  false, false);
    }

    // D layout: VGPR v, lane 0  -> row M=v   (N=0);
    //                   lane 16 -> row M=v+8 (N=0). All other lanes hold 0,
    // so their contributions are exactly 0 -> no masking needed.
#pragma unroll
    for (int v = 0; v < 8; ++v) {
      float sp = dS1p[v], st = dS1t[v], sd = dSd[v];
      float q    = (sp - st) * (sp + st);          // S1p^2 - S1t^2
      float diff = fmaf(q, -inv_n, sd) * inv_nm1;  // var_p - var_t
      acc = fmaf(diff, diff, acc);
    }
  }

  acc *= valid;

  // Fixed-order block reduction (deterministic; no float atomics anywhere).
  s_red[tid] = acc;
  __syncthreads();
  for (int s = 128; s > 0; s >>= 1) {
    if (tid < s) s_red[tid] += s_red[tid + s];
    __syncthreads();
  }
  if (tid == 0) partial[bid] = s_red[0];
}

// ---------------------------------------------------------------------------
// Kernel 2: deterministic final reduction of the 2048 partials + n_multi.
// ---------------------------------------------------------------------------
__global__ __launch_bounds__(256) void rc_finalize_kernel(
    const float* __restrict__ partial, const float* __restrict__ mask,
    float* __restrict__ out)
{
  __shared__ float s_sum[256];
  __shared__ float s_cnt[256];
  const int tid = threadIdx.x;

  float s = 0.f;
  for (int i = tid; i < NB; i += 256) s += partial[i];

  float c = 0.f;
  for (int b = tid; b < B_DIM; b += 256) {
    float nb = 0.f;
    for (int r = 0; r < R_DIM; ++r) nb += mask[b * R_DIM + r];
    c += (nb > 1.0f) ? 1.0f : 0.0f;
  }

  s_sum[tid] = s;
  s_cnt[tid] = c;
  __syncthreads();
  for (int st = 128; st > 0; st >>= 1) {
    if (tid < st) {
      s_sum[tid] += s_sum[tid + st];
      s_cnt[tid] += s_cnt[tid + st];
    }
    __syncthreads();
  }
  if (tid == 0) {
    float total = s_sum[0];
    float nm    = s_cnt[0];
    out[0] = (nm > 0.f) ? (0.1f * total / ((float)T_DIM * nm)) : 0.0f;
  }
}

extern "C" void kernel_launch(void* const* d_in, const int* in_sizes, int n_in,
                              void* d_out, int out_size, void* d_ws, size_t ws_size,
                              hipStream_t stream) {
  const float* pred    = (const float*)d_in[0];  // (512, 4096, 16) f32
  const float* target  = (const float*)d_in[1];  // (512, 4096, 16) f32
  const float* mask    = (const float*)d_in[2];  // (512, 16) f32
  float*       out     = (float*)d_out;          // scalar f32
  float*       partial = (float*)d_ws;           // NB floats (8 KB) scratch

  rc_var_wmma_kernel<<<NB, 256, 0, stream>>>(pred, target, mask, partial);
  rc_finalize_kernel<<<1, 256, 0, stream>>>(partial, mask, out);
}